// DifferentialMaxtree_23373212024968
// MI455X (gfx1250) — compile-verified
//
#include <hip/hip_runtime.h>

#define HN (2048 * 2048)          // number of max-tree nodes
#define EPSF 1e-10f
#define DOUBLING_STEPS 23         // ceil(log2(2^22)) + 1, matches reference

typedef __attribute__((ext_vector_type(2))) float v2f;
typedef __attribute__((ext_vector_type(8))) float v8f;

// ---------------------------------------------------------------------------
// Phase 1: features -> WMMA dot(17) -> sigmoid -> contrib, packed with parent
// One wave = 16 nodes. Lane l: node = base + (l&15), half = l>>4 selects which
// K-pair of each 4-wide chunk this lane supplies (A and B use the same
// (vgpr, half) -> K mapping, so w_k pairs with f_k).
// ---------------------------------------------------------------------------
__global__ __launch_bounds__(256) void phase1_contrib(
    const int* __restrict__ parent,
    const float* __restrict__ diff,
    const float* __restrict__ attr,     // (N,15) row-major
    const float* __restrict__ weight,   // 17 floats
    const float* __restrict__ bias,     // 1 float
    unsigned long long* __restrict__ buf0)  // packed {c:f32 lo, p:i32 hi}, N+1 entries
{
    const int lane = threadIdx.x & 31;
    const int half = lane >> 4;                 // 0 or 1
    const int m    = lane & 15;                 // node within tile
    const int wave = blockIdx.x * (blockDim.x >> 5) + (threadIdx.x >> 5);
    const int base = wave << 4;
    const int row  = base + m;
    const float* a = attr + (size_t)row * 15;

    // Per-lane B-fragment values for 5 chunks: feature k = 4c + 2*half (+1)
    // k0..4 = a0..a4 ; k5..13 = log(|a6..a14|+eps) ; k14 = lshape ; k15 = cos(a5) ; k16 = sin(a5)
    float bx[5], by[5];
    if (half == 0) {
        // k = {0,1},{4,5},{8,9},{12,13},{16,17}
        float a5 = a[5], a6 = a[6];
        bx[0] = a[0];                        by[0] = a[1];
        bx[1] = a[4];                        by[1] = logf(fabsf(a6) + EPSF);
        bx[2] = logf(fabsf(a[9])  + EPSF);   by[2] = logf(fabsf(a[10]) + EPSF);
        bx[3] = logf(fabsf(a[13]) + EPSF);   by[3] = logf(fabsf(a[14]) + EPSF);
        bx[4] = sinf(a5);                    by[4] = 0.0f;   // k17 pad
    } else {
        // k = {2,3},{6,7},{10,11},{14,15},{18,19}
        float a5 = a[5], a6 = a[6], a7 = a[7];
        bx[0] = a[2];                        by[0] = a[3];
        bx[1] = logf(fabsf(a7)    + EPSF);   by[1] = logf(fabsf(a[8])  + EPSF);
        bx[2] = logf(fabsf(a[11]) + EPSF);   by[2] = logf(fabsf(a[12]) + EPSF);
        bx[3] = sqrtf(a7) / (sqrtf(a6) + EPSF);  by[3] = cosf(a5);   // lshape, cos
        bx[4] = 0.0f;                        by[4] = 0.0f;   // k18,k19 pad
    }

    // A-fragment: weight chunk broadcast over all 16 rows (same lane->K map)
    float ax[5], ay[5];
#pragma unroll
    for (int c = 0; c < 5; ++c) {
        int kx = 4 * c + 2 * half;
        int ky = kx + 1;
        ax[c] = (kx < 17) ? weight[kx] : 0.0f;
        ay[c] = (ky < 17) ? weight[ky] : 0.0f;
    }

    // D[m][n] = sum_k w_k * feats[node n][k]; rows identical, every lane's
    // accumulator holds the dot product of node (base + lane%16).
    v8f acc = {};
#pragma unroll
    for (int c = 0; c < 5; ++c) {
        v2f A; A.x = ax[c]; A.y = ay[c];
        v2f B; B.x = bx[c]; B.y = by[c];
        acc = __builtin_amdgcn_wmma_f32_16x16x4_f32(
            /*neg_a=*/false, A, /*neg_b=*/false, B,
            /*c_mod=*/(short)0, acc, /*reuse_a=*/false, /*reuse_b=*/false);
    }

    float val   = acc[0] + bias[0];
    float score = 1.0f / (1.0f + expf(-val));

    if (half == 0) {
        float contrib = diff[row] * score;
        unsigned long long packed =
            ((unsigned long long)(unsigned int)parent[row] << 32) |
            (unsigned long long)__float_as_uint(contrib);
        buf0[row] = packed;   // coalesced 16-lane b64 store
    }
    if (blockIdx.x == 0 && threadIdx.x == 0) {
        // sentinel virtual root: c = 0.0f, p = N (self-loop under doubling)
        buf0[HN] = ((unsigned long long)(unsigned int)HN << 32);
    }
}

// ---------------------------------------------------------------------------
// Phase 2: one pointer-doubling step on packed {c,p}. Single b64 gather per
// node (both c[p] and p[p] in one cacheline hit); 64MB working set is
// L2-resident on MI455X (192MB L2).
// ---------------------------------------------------------------------------
__global__ __launch_bounds__(256) void doubling_step(
    const unsigned long long* __restrict__ in,
    unsigned long long* __restrict__ out,
    int n)  // n = N+1 (includes sentinel)
{
    int i = blockIdx.x * blockDim.x + threadIdx.x;
    if (i >= n) return;
    unsigned long long v = in[i];
    int p   = (int)(v >> 32);
    float c = __uint_as_float((unsigned int)v);
    unsigned long long w = in[p];                       // single b64 gather
    float cp = __uint_as_float((unsigned int)w);
    out[i] = (w & 0xFFFFFFFF00000000ull)                // p' = p[p]
           | (unsigned long long)__float_as_uint(c + cp); // c' = c + c[p]
}

// Final (23rd) step: compute c + c[p] and emit float result only.
__global__ __launch_bounds__(256) void final_step(
    const unsigned long long* __restrict__ in,
    float* __restrict__ out,
    int n)  // n = N
{
    int i = blockIdx.x * blockDim.x + threadIdx.x;
    if (i >= n) return;
    unsigned long long v = in[i];
    int p   = (int)(v >> 32);
    float c = __uint_as_float((unsigned int)v);
    unsigned long long w = in[p];
    out[i] = c + __uint_as_float((unsigned int)w);
}

// ---------------------------------------------------------------------------
extern "C" void kernel_launch(void* const* d_in, const int* in_sizes, int n_in,
                              void* d_out, int out_size, void* d_ws, size_t ws_size,
                              hipStream_t stream) {
    const int*   parent = (const int*)d_in[0];
    const float* diff   = (const float*)d_in[1];
    const float* attr   = (const float*)d_in[2];
    const float* weight = (const float*)d_in[3];
    const float* bias   = (const float*)d_in[4];
    float* out = (float*)d_out;

    unsigned long long* buf0 = (unsigned long long*)d_ws;
    unsigned long long* buf1 = buf0 + (HN + 1);

    // Phase 1: 16 nodes per wave, 8 waves per block
    {
        int blocks = HN / (16 * 8);   // 32768, exact
        phase1_contrib<<<blocks, 256, 0, stream>>>(parent, diff, attr, weight, bias, buf0);
    }

    // Phase 2: 22 ping-pong steps + fused final step = 23 total (reference-exact)
    {
        int n = HN + 1;
        int blocks = (n + 255) / 256;
        unsigned long long* a = buf0;
        unsigned long long* b = buf1;
        for (int s = 0; s < DOUBLING_STEPS - 1; ++s) {
            doubling_step<<<blocks, 256, 0, stream>>>(a, b, n);
            unsigned long long* t = a; a = b; b = t;
        }
        // 22 swaps (even) -> result back in buf0
        final_step<<<(HN + 255) / 256, 256, 0, stream>>>(a, out, HN);
    }
}